// HeteroRGCN_24429773980175
// MI455X (gfx1250) — compile-verified
//
#include <hip/hip_runtime.h>
#include <hip/hip_bf16.h>

// HeteroRGCN on gfx1250:
//   layer(h): per-etype  Wh_e = h @ W_e + b_e   (WMMA f32 16x16x4 GEMM)
//             msg mean per dst per etype, summed over etypes (atomic f32 scatter)
//   out = layer2( leaky_relu( layer1(x) ) )
//
// Sizes: N=100000 nodes, E=200000 edges/etype, 3 etypes, IN=64, HID=OUT=128.

typedef __attribute__((ext_vector_type(2))) float v2f;
typedef __attribute__((ext_vector_type(8))) float v8f;

#define RGCN_ETYPES 3
#define RGCN_IN     64
#define RGCN_H      128

// ---------------------------------------------------------------- zero fill
__global__ void rgcn_zero_kernel(float4* __restrict__ p, long n4) {
    long i = (long)blockIdx.x * blockDim.x + threadIdx.x;
    if (i < n4) p[i] = make_float4(0.f, 0.f, 0.f, 0.f);
}

// ------------------------------------------------------------ degree count
// deg is an int view of the rdeg buffer (zero-initialized beforehand).
__global__ void rgcn_count_kernel(int* __restrict__ deg, const int* __restrict__ dst,
                                  int N, int E, int total) {
    int i = blockIdx.x * blockDim.x + threadIdx.x;
    if (i >= total) return;
    int et = i / E;
    int d  = dst[i];
    atomicAdd(&deg[et * N + d], 1);
}

// ------------------------------------------------- int count -> reciprocal
__global__ void rgcn_recip_kernel(float* __restrict__ rdeg, int n) {
    int i = blockIdx.x * blockDim.x + threadIdx.x;
    if (i >= n) return;
    int v = reinterpret_cast<const int*>(rdeg)[i];
    rdeg[i] = (v > 0) ? (1.0f / (float)v) : 0.0f;
}

// ----------------------------------------------------------- WMMA GEMM
// C[m][n] = sum_k A[m][k] * W[k][n] + bias[n], A:[nrows x K], W:[K x 128].
// Block = 128 threads = 4 waves; each wave computes a 16x128 output strip
// as 8 accumulators of V_WMMA_F32_16X16X4_F32.
//
// W is staged through LDS in 64-row K chunks in a K-PAIR-INTERLEAVED layout:
//   ldsW[(k/2)*PSTR + col*2 + (k&1)] = W[k][col]
// so each B fragment (W[k+koff][col], W[k+koff+1][col]) is ONE aligned
// ds_load_b64 landing directly in an even VGPR pair (no mov shuffling).
// PSTR = 288 floats => half-wave row offset is 288 mod 64 = 32 banks:
// lanes 0-15 cover banks b..b+31, lanes 16-31 cover b+32..b+63 (no conflicts).
template <int K, bool LEAKY>
__global__ __launch_bounds__(128)
void rgcn_gemm_kernel(const float* __restrict__ A, const float* __restrict__ W,
                      const float* __restrict__ bias, float* __restrict__ outp,
                      int nrows) {
    constexpr int CK   = 64;    // K-chunk staged in LDS
    constexpr int PSTR = 288;   // floats per k-pair row: 2*128 data + 32 pad
    __shared__ float ldsW[(CK / 2) * PSTR];   // 36 KB

    const int tid  = threadIdx.x;
    const int wave = tid >> 5;
    const int lane = tid & 31;
    const int l15  = lane & 15;
    const int hi   = lane >> 4;        // 0: lanes 0-15, 1: lanes 16-31
    const int koff = hi * 2;           // A fragment K sub-offset per ISA layout
    const int m0   = (blockIdx.x * 4 + wave) * 16;
    const bool valid = (m0 < nrows);
    const long arow  = valid ? (long)(m0 + l15) : 0;   // clamp OOB waves to row 0
    const float* Ap  = A + arow * K;

    v8f acc[8] = {};

    for (int kb = 0; kb < K; kb += CK) {
        __syncthreads();   // WAR guard vs previous chunk's ds reads
        // cooperative stage of W[kb .. kb+CK) x 128 into pair-interleaved LDS
        const float4* W4 = reinterpret_cast<const float4*>(W + (long)kb * RGCN_H);
        for (int j = tid; j < CK * (RGCN_H / 4); j += 128) {
            int kk = j >> 5;           // row within chunk
            int cc = (j & 31) << 2;    // column (multiple of 4)
            float4 w = W4[j];
            float* dp = &ldsW[(kk >> 1) * PSTR + cc * 2 + (kk & 1)];
            dp[0] = w.x; dp[2] = w.y; dp[4] = w.z; dp[6] = w.w;
        }
        __syncthreads();

        // per-lane LDS base: k-pair (k/2 + hi), column pair at l15*2
        const float* wbase = &ldsW[hi * PSTR + l15 * 2];
        for (int k = 0; k < CK; k += 4) {
            // A fragment (16x4 f32): lanes 0-15 -> K=k,k+1 ; lanes 16-31 -> K=k+2,k+3
            v2f a = *reinterpret_cast<const v2f*>(Ap + kb + k + koff);
            if (LEAKY) {
                a.x = (a.x > 0.0f) ? a.x : a.x * 0.01f;
                a.y = (a.y > 0.0f) ? a.y : a.y * 0.01f;
            }
            const float* wk = wbase + (k >> 1) * PSTR;
#pragma unroll
            for (int nt = 0; nt < 8; ++nt) {
                // B fragment: one b64 = (W[k+koff][col], W[k+koff+1][col])
                v2f b = *reinterpret_cast<const v2f*>(wk + nt * 32);
                acc[nt] = __builtin_amdgcn_wmma_f32_16x16x4_f32(
                    false, a, false, b, (short)0, acc[nt], false, false);
            }
        }
    }

    if (!valid) return;   // after all barriers

    // D layout: vgpr r holds row (m0 + r + 8*hi), col = nt*16 + l15
    const int rbase = m0 + hi * 8;
#pragma unroll
    for (int nt = 0; nt < 8; ++nt) {
        const int col = nt * 16 + l15;
        const float bv = bias[col];
#pragma unroll
        for (int r = 0; r < 8; ++r) {
            const int row = rbase + r;
            if (row < nrows)
                outp[(long)row * RGCN_H + col] = acc[nt][r] + bv;
        }
    }
}

// -------------------------------------------------------- edge scatter-mean
// One wave per edge; each lane handles one float4 of the 128-wide feature.
// out[dst] += Wh[src] * rdeg[dst]   (native global_atomic_add_f32, no return)
__global__ void rgcn_scatter_kernel(const float* __restrict__ Wh,
                                    const int* __restrict__ src,
                                    const int* __restrict__ dst,
                                    const float* __restrict__ rdeg,
                                    float* __restrict__ outp, int nE) {
    int gid = blockIdx.x * blockDim.x + threadIdx.x;
    int e   = gid >> 5;
    int c   = gid & 31;
    if (e >= nE) return;
    int s = src[e];
    int d = dst[e];
    float r = rdeg[d];
    float4 v = reinterpret_cast<const float4*>(Wh)[(long)s * (RGCN_H / 4) + c];
    float* o = outp + (long)d * RGCN_H + c * 4;
    unsafeAtomicAdd(o + 0, v.x * r);
    unsafeAtomicAdd(o + 1, v.y * r);
    unsafeAtomicAdd(o + 2, v.z * r);
    unsafeAtomicAdd(o + 3, v.w * r);
}

// ---------------------------------------------------------------- launcher
extern "C" void kernel_launch(void* const* d_in, const int* in_sizes, int n_in,
                              void* d_out, int out_size, void* d_ws, size_t ws_size,
                              hipStream_t stream) {
    const float* x    = (const float*)d_in[0];   // [N, 64]
    const int*   esrc = (const int*)  d_in[1];   // [3, E]
    const int*   edst = (const int*)  d_in[2];   // [3, E]
    const float* W1   = (const float*)d_in[3];   // [3, 64, 128]
    const float* b1   = (const float*)d_in[4];   // [3, 128]
    const float* W2   = (const float*)d_in[5];   // [3, 128, 128]
    const float* b2   = (const float*)d_in[6];   // [3, 128]
    float* out = (float*)d_out;                  // [N, 128]

    const int N = in_sizes[0] / RGCN_IN;         // 100000
    const int E = in_sizes[1] / RGCN_ETYPES;     // 200000

    // Workspace layout (floats): rdeg[3N] | h1[N*128] | Wh temp[N*128]  (~104 MB)
    float* rdeg = (float*)d_ws;
    float* h1   = rdeg + (size_t)RGCN_ETYPES * N;
    float* Wh   = h1 + (size_t)N * RGCN_H;

    const int TB = 256;
    auto blocks = [](long n, int tb) { return (unsigned)((n + tb - 1) / tb); };

    // 1) zero rdeg (doubles as int deg) + h1
    {
        long n4 = ((long)RGCN_ETYPES * N + (long)N * RGCN_H) / 4;
        rgcn_zero_kernel<<<blocks(n4, TB), TB, 0, stream>>>((float4*)rdeg, n4);
    }
    // 2) degree counts (int atomics) then reciprocal in place
    {
        int total = RGCN_ETYPES * E;
        rgcn_count_kernel<<<blocks(total, TB), TB, 0, stream>>>((int*)rdeg, edst, N, E, total);
        int n = RGCN_ETYPES * N;
        rgcn_recip_kernel<<<blocks(n, TB), TB, 0, stream>>>(rdeg, n);
    }

    const unsigned gemm_blocks    = (unsigned)((N + 63) / 64);
    const unsigned scatter_blocks = blocks((long)E * 32, TB);

    // 3) layer 1: per-etype GEMM + scatter-mean into h1
    for (int e = 0; e < RGCN_ETYPES; ++e) {
        rgcn_gemm_kernel<RGCN_IN, false><<<gemm_blocks, 128, 0, stream>>>(
            x, W1 + (size_t)e * RGCN_IN * RGCN_H, b1 + (size_t)e * RGCN_H, Wh, N);
        rgcn_scatter_kernel<<<scatter_blocks, TB, 0, stream>>>(
            Wh, esrc + (size_t)e * E, edst + (size_t)e * E, rdeg + (size_t)e * N, h1, E);
    }

    // 4) zero d_out, then layer 2 (leaky-ReLU fused into GEMM A loads)
    {
        long n4 = (long)N * RGCN_H / 4;
        rgcn_zero_kernel<<<blocks(n4, TB), TB, 0, stream>>>((float4*)out, n4);
    }
    for (int e = 0; e < RGCN_ETYPES; ++e) {
        rgcn_gemm_kernel<RGCN_H, true><<<gemm_blocks, 128, 0, stream>>>(
            h1, W2 + (size_t)e * RGCN_H * RGCN_H, b2 + (size_t)e * RGCN_H, Wh, N);
        rgcn_scatter_kernel<<<scatter_blocks, TB, 0, stream>>>(
            Wh, esrc + (size_t)e * E, edst + (size_t)e * E, rdeg + (size_t)e * N, out, E);
    }
}